// PatchTransformer_89223650607105
// MI455X (gfx1250) — compile-verified
//
#include <hip/hip_runtime.h>
#include <cstdint>

#define IMG 512
#define PS  128
#define NB  8
#define ROWS_PER_BLOCK 32
#define THREADS 256

__global__ __launch_bounds__(THREADS) void patch_transformer_fused(
    const float* __restrict__ patch,   // [3,128,128]
    const int*   __restrict__ boxes,   // [32,8,4]
    const float* __restrict__ angles,  // [32]
    const float* __restrict__ trans,   // [32,2]
    const float* __restrict__ scales,  // [32]
    float*       __restrict__ out)     // [32,3,512,512]
{
    __shared__ float         s_patch[PS * PS];     // 64 KB: one channel
    __shared__ unsigned char s_rowmask[IMG];       // bit n: row covered by box n
    __shared__ unsigned char s_colmask[IMG];
    __shared__ int           s_x1[NB];
    __shared__ int           s_y1[NB];
    __shared__ float         s_scl[NB];

    const int blk  = blockIdx.x;
    const int tile = blk & 15;          // 16 row tiles of 32 rows
    const int ch   = (blk >> 4) % 3;
    const int b    = blk / 48;
    const int tid  = threadIdx.x;

    // ---- CDNA5 async DMA: stage patch channel into LDS (ASYNCcnt path) ----
    {
        const float* src = patch + ch * PS * PS;
        #pragma unroll
        for (int k = 0; k < (PS * PS) / (4 * THREADS); ++k) {   // 16 x b128 per lane
            int idx4 = k * THREADS + tid;                        // float4 index
            uint32_t lds_off = (uint32_t)(uintptr_t)(&s_patch[idx4 * 4]);
            const float* g = src + idx4 * 4;
            asm volatile("global_load_async_to_lds_b128 %0, %1, off"
                         :: "v"(lds_off), "v"(g) : "memory");
        }
    }

    // ---- box params + coverage masks (overlaps with async DMA in flight) ----
    int   bx1[NB], by1[NB], bsz[NB];
    float bscl[NB];
    #pragma unroll
    for (int n = 0; n < NB; ++n) {
        const int* bp = boxes + (b * NB + n) * 4;
        int x0 = min(bp[0], IMG), y0 = min(bp[1], IMG);
        int x2 = min(bp[2], IMG), y2 = min(bp[3], IMG);
        int bw = x2 - x0, bh = y2 - y0;
        int sz = (min(bw, bh) / 2) * 2;                 // PORTION == 1
        bx1[n] = bw / 2 - sz / 2;                       // faithful: no box offset added
        by1[n] = bh / 2 - sz / 2;
        bsz[n] = sz;
        bscl[n] = 128.0f / (float)max(sz, 1);
    }
    if (tid < NB) { s_x1[tid] = bx1[tid]; s_y1[tid] = by1[tid]; s_scl[tid] = bscl[tid]; }
    #pragma unroll
    for (int k = 0; k < IMG / THREADS; ++k) {
        int i = k * THREADS + tid;
        unsigned rm = 0, cm = 0;
        #pragma unroll
        for (int n = 0; n < NB; ++n) {
            if (bsz[n] > 0) {
                if (i >= bx1[n] && i < bx1[n] + bsz[n]) rm |= 1u << n;
                if (i >= by1[n] && i < by1[n] + bsz[n]) cm |= 1u << n;
            }
        }
        s_rowmask[i] = (unsigned char)rm;
        s_colmask[i] = (unsigned char)cm;
    }

    asm volatile("s_wait_asynccnt 0" ::: "memory");     // DMA complete
    __syncthreads();                                    // masks + patch visible

    // ---- per-image affine params ----
    const float t0 = trans[b * 2 + 0];
    const float t1 = trans[b * 2 + 1];
    const float s  = scales[b];
    const float inv_s = 1.0f / s;                       // hoist division out of hot loop
    const float th = angles[b] * 0.017453292519943295f;
    const float cosv = cosf(th);
    const float sinv = sinf(th);
    const float cc = 255.5f;                            // (H-1)/2

    // coverage-mask fetch with implicit bounds check (OOB -> 0 -> zero fill)
    auto rowm = [&](int R) -> unsigned {
        return ((unsigned)R < (unsigned)IMG) ? (unsigned)s_rowmask[R] : 0u;
    };
    auto colm = [&](int C) -> unsigned {
        return ((unsigned)C < (unsigned)IMG) ? (unsigned)s_colmask[C] : 0u;
    };

    // canvas value at (row R, col C) given combined coverage mask m
    // (m != 0 implies R,C in range, since masks are only set for valid indices)
    auto canvas_at = [&](int R, int C, unsigned m) -> float {
        if (m == 0u) return 0.0f;                       // dominant fast path
        int n = 31 - __clz((int)m);                     // last (highest) box wins
        float scl = s_scl[n];
        float sr = ((float)(R - s_x1[n]) + 0.5f) * scl - 0.5f;
        float sc = ((float)(C - s_y1[n]) + 0.5f) * scl - 0.5f;
        float fr = floorf(sr), fc = floorf(sc);
        float wr = sr - fr,   wc = sc - fc;
        int r0 = (int)fr, c0 = (int)fc;
        int r0c = min(max(r0,     0), PS - 1);
        int r1c = min(max(r0 + 1, 0), PS - 1);
        int c0c = min(max(c0,     0), PS - 1);
        int c1c = min(max(c0 + 1, 0), PS - 1);
        const float* p0 = &s_patch[r0c * PS];
        const float* p1 = &s_patch[r1c * PS];
        float v00 = p0[c0c], v01 = p0[c1c], v10 = p1[c0c], v11 = p1[c1c];
        return v00 * (1.0f - wr) * (1.0f - wc) + v01 * (1.0f - wr) * wc
             + v10 * wr * (1.0f - wc)          + v11 * wr * wc;
    };

    const int row0 = tile * ROWS_PER_BLOCK;
    float* outp = out + (size_t)(b * 3 + ch) * IMG * IMG;

    for (int rr = 0; rr < ROWS_PER_BLOCK; ++rr) {
        int r = row0 + rr;
        float rowv = (float)r - cc - t1;
        #pragma unroll
        for (int k = 0; k < IMG / THREADS; ++k) {
            int q = k * THREADS + tid;
            float colv  = (float)q - cc - t0;
            float src_c = ( cosv * colv + sinv * rowv) * inv_s + cc;
            float src_r = (-sinv * colv + cosv * rowv) * inv_s + cc;
            float fr = floorf(src_r), fc = floorf(src_c);
            int   R0 = (int)fr,       C0 = (int)fc;
            float wr = src_r - fr,    wc = src_c - fc;

            // 2x2 stencil shares mask rows/cols: 4 DS byte loads instead of 8
            unsigned rm0 = rowm(R0), rm1 = rowm(R0 + 1);
            unsigned cm0 = colm(C0), cm1 = colm(C0 + 1);

            float v00 = canvas_at(R0,     C0,     rm0 & cm0);
            float v01 = canvas_at(R0,     C0 + 1, rm0 & cm1);
            float v10 = canvas_at(R0 + 1, C0,     rm1 & cm0);
            float v11 = canvas_at(R0 + 1, C0 + 1, rm1 & cm1);

            float res = v00 * (1.0f - wr) * (1.0f - wc) + v01 * (1.0f - wr) * wc
                      + v10 * wr * (1.0f - wc)          + v11 * wr * wc;
            // write-once 100 MB output: bypass cache retention
            __builtin_nontemporal_store(res, &outp[r * IMG + q]);
        }
    }
}

extern "C" void kernel_launch(void* const* d_in, const int* in_sizes, int n_in,
                              void* d_out, int out_size, void* d_ws, size_t ws_size,
                              hipStream_t stream) {
    (void)in_sizes; (void)n_in; (void)out_size; (void)d_ws; (void)ws_size;
    const float* patch  = (const float*)d_in[0];
    const int*   boxes  = (const int*)  d_in[1];
    const float* angles = (const float*)d_in[2];
    const float* trans  = (const float*)d_in[3];
    const float* scales = (const float*)d_in[4];
    float* out = (float*)d_out;

    dim3 grid(32 * 3 * (IMG / ROWS_PER_BLOCK));   // 1536 blocks
    dim3 block(THREADS);                           // 8 wave32 waves
    patch_transformer_fused<<<grid, block, 0, stream>>>(patch, boxes, angles,
                                                        trans, scales, out);
}